// OhemCELoss_63273458204677
// MI455X (gfx1250) — compile-verified
//
#include <hip/hip_runtime.h>
#include <hip/hip_bf16.h>
#include <math.h>

typedef __attribute__((ext_vector_type(2))) float v2f;
typedef __attribute__((ext_vector_type(8))) float v8f;

#define NCLS     19
#define IMG_HW   (768 * 768)                 // 589824 pixels per image
#define NPIX     (8LL * IMG_HW)              // 4,718,592 pixels total
#define MIN_KEPT 131072
#define THRESH   0.9f
#define FBLOCKS  1024

// ---- workspace layout (bytes) ----
// [0)      ctrl: 4 x u32  : [0]=n_valid  [1]=remaining_k  [2]=key prefix  [3]=threshold (float)
// [64)     hist: 4 levels x 256 x u32
// [4608)   per-block partial loss: FBLOCKS doubles
// [12800)  per-block partial count: FBLOCKS ints
// [32768)  logp array: NPIX floats (18.9 MB; fits in 192 MB L2)
#define WS_CTRL  0
#define WS_HIST  64
#define WS_PLOSS 4608
#define WS_PCNT  12800
#define WS_LOGP  32768

// Order-preserving map float -> u32 (valid for all finite/inf values)
__device__ __forceinline__ unsigned keyOf(float f) {
  unsigned b = __float_as_uint(f);
  return (b & 0x80000000u) ? ~b : (b ^ 0x80000000u);
}

__global__ void k_init(unsigned* ws) {
  const int stride = gridDim.x * blockDim.x;
  for (int j = blockIdx.x * blockDim.x + threadIdx.x; j < 4224; j += stride)
    ws[j] = 0u;  // zero bytes [0, 16896)
}

// Pass 1: log-softmax gather + level-0 key histogram + n_valid.
// Wave layout: pixel = waveId*16 + (lane&15); half = lane>>4 owns channels 4j+2*half+{0,1}.
// Sum-of-exp is computed on the matrix unit: D = ones(16x4) x B(4x16) accumulated over 5
// K-chunks => every lane's D[0] holds the column (=pixel) sum of exp.
__global__ void k_pass1(const float* __restrict__ pred, const int* __restrict__ tgt,
                        float* __restrict__ logp, unsigned* __restrict__ hist0,
                        unsigned* __restrict__ ctrl) {
  __shared__ unsigned lh[256];
  __shared__ unsigned lvalid;
  lh[threadIdx.x & 255] = 0u;
  if (threadIdx.x == 0) lvalid = 0u;
  __syncthreads();

  const int lane = threadIdx.x & 31;
  const int half = lane >> 4;
  const int col  = lane & 15;
  const long long wave = (long long)blockIdx.x * (blockDim.x >> 5) + (threadIdx.x >> 5);
  const long long pix  = wave * 16 + col;

  const int n  = (int)(pix / IMG_HW);
  const int hw = (int)(pix % IMG_HW);
  const float* pp = pred + (long long)n * NCLS * IMG_HW + hw;
  const int t = tgt[pix];

  // Load this lane's 10 channel slots (channel 19 slot padded with -inf).
  float x[5][2];
#pragma unroll
  for (int j = 0; j < 5; ++j) {
#pragma unroll
    for (int u = 0; u < 2; ++u) {
      const int ch = 4 * j + 2 * half + u;
      x[j][u] = (ch < NCLS)
                    ? __builtin_nontemporal_load(pp + (long long)ch * IMG_HW)
                    : -INFINITY;
    }
  }

  // channel max (pairwise across half-waves)
  float m = x[0][0];
#pragma unroll
  for (int j = 0; j < 5; ++j) {
    m = fmaxf(m, x[j][0]);
    m = fmaxf(m, x[j][1]);
  }
  m = fmaxf(m, __shfl_xor(m, 16, 32));

  // target logit (exactly one lane of the pair owns channel tt)
  const int tt = (t < 0) ? 0 : t;
  float xt = -INFINITY;
#pragma unroll
  for (int j = 0; j < 5; ++j) {
#pragma unroll
    for (int u = 0; u < 2; ++u) {
      if (4 * j + 2 * half + u == tt) xt = x[j][u];
    }
  }
  xt = fmaxf(xt, __shfl_xor(xt, 16, 32));

  // sum(exp(x-m)) via V_WMMA_F32_16X16X4_F32, A = ones
  v2f a;
  a.x = 1.0f;
  a.y = 1.0f;
  v8f acc = {0.f, 0.f, 0.f, 0.f, 0.f, 0.f, 0.f, 0.f};
#pragma unroll
  for (int j = 0; j < 5; ++j) {
    v2f b;
    b.x = expf(x[j][0] - m);  // exp(-inf) == 0 for the padded slot
    b.y = expf(x[j][1] - m);
    acc = __builtin_amdgcn_wmma_f32_16x16x4_f32(false, a, false, b, (short)0, acc,
                                                false, false);
  }
  const float sumexp = acc[0];

  float lp = (xt - m) - logf(sumexp);
  if (t < 0) lp = INFINITY;  // sentinel: sorts past all valid keys, never selected

  if (half == 0) {
    logp[pix] = lp;
    atomicAdd(&lh[keyOf(lp) >> 24], 1u);
    if (t >= 0) atomicAdd(&lvalid, 1u);
  }
  __syncthreads();
  const unsigned c = lh[threadIdx.x & 255];
  if (threadIdx.x < 256 && c) atomicAdd(&hist0[threadIdx.x], c);
  if (threadIdx.x == 0 && lvalid) atomicAdd(&ctrl[0], lvalid);
}

// Radix-select refinement histogram for byte `level` (1..3). logp is L2-resident.
__global__ void k_rhist(const float* __restrict__ logp, const unsigned* __restrict__ ctrl,
                        unsigned* __restrict__ hist, int level) {
  __shared__ unsigned lh[256];
  lh[threadIdx.x & 255] = 0u;
  __syncthreads();
  const unsigned prefix = ctrl[2];
  const int msh = 32 - 8 * level;
  const int bsh = 24 - 8 * level;
  const long long stride = (long long)gridDim.x * blockDim.x;
  for (long long i = (long long)blockIdx.x * blockDim.x + threadIdx.x; i < NPIX; i += stride) {
    const unsigned key = keyOf(logp[i]);
    if (((key ^ prefix) >> msh) == 0)
      atomicAdd(&lh[(key >> bsh) & 255u], 1u);
  }
  __syncthreads();
  const unsigned c = lh[threadIdx.x & 255];
  if (threadIdx.x < 256 && c) atomicAdd(&hist[threadIdx.x], c);
}

// Scan one 256-bin histogram, advance (prefix, k). level==0 also derives k from n_valid;
// level==3 finalizes threshold = max(exp(logp_kth), 0.9).
__global__ void k_scan(unsigned* ctrl, const unsigned* __restrict__ hist, int level) {
  if (threadIdx.x != 0 || blockIdx.x != 0) return;
  long long k;
  unsigned prefix;
  if (level == 0) {
    long long kk = (long long)(int)ctrl[0] - 1;
    if (kk > MIN_KEPT) kk = MIN_KEPT;
    if (kk < 0) kk = 0;
    k = kk;
    prefix = 0u;
  } else {
    k = (long long)(int)ctrl[1];
    prefix = ctrl[2];
  }
  unsigned long long cum = 0;
  int b = 255;
  for (int i = 0; i < 256; ++i) {
    const unsigned long long c = hist[i];
    if (cum + c > (unsigned long long)k) { b = i; break; }
    cum += c;
  }
  k -= (long long)cum;
  prefix |= ((unsigned)b) << (24 - 8 * level);
  ctrl[1] = (unsigned)k;
  ctrl[2] = prefix;
  if (level == 3) {
    const unsigned bits = (prefix & 0x80000000u) ? (prefix ^ 0x80000000u) : ~prefix;
    const float lpk = __uint_as_float(bits);
    ((float*)ctrl)[3] = fmaxf(expf(lpk), THRESH);
  }
}

// Final selection + deterministic two-stage reduction (no float atomics).
__global__ void k_final(const float* __restrict__ logp, const int* __restrict__ tgt,
                        const float* __restrict__ w, const unsigned* __restrict__ ctrl,
                        double* __restrict__ pLoss, int* __restrict__ pCnt) {
  __shared__ double sl[256];
  __shared__ int sc[256];
  const float thr = ((const float*)ctrl)[3];
  double ls = 0.0;
  int cnt = 0;
  const long long stride = (long long)gridDim.x * blockDim.x;
  for (long long i = (long long)blockIdx.x * blockDim.x + threadIdx.x; i < NPIX; i += stride) {
    const int t = tgt[i];
    if (t >= 0) {
      const float lp = logp[i];
      if (expf(lp) < thr) {
        ls += (double)(-w[t] * lp);
        ++cnt;
      }
    }
  }
  sl[threadIdx.x] = ls;
  sc[threadIdx.x] = cnt;
  __syncthreads();
  for (int s = 128; s > 0; s >>= 1) {
    if (threadIdx.x < s) {
      sl[threadIdx.x] += sl[threadIdx.x + s];
      sc[threadIdx.x] += sc[threadIdx.x + s];
    }
    __syncthreads();
  }
  if (threadIdx.x == 0) {
    pLoss[blockIdx.x] = sl[0];
    pCnt[blockIdx.x]  = sc[0];
  }
}

__global__ void k_fin(const double* __restrict__ pLoss, const int* __restrict__ pCnt,
                      float* __restrict__ out) {
  __shared__ double sl[256];
  __shared__ long long sc[256];
  double ls = 0.0;
  long long cnt = 0;
  for (int i = threadIdx.x; i < FBLOCKS; i += 256) {
    ls += pLoss[i];
    cnt += pCnt[i];
  }
  sl[threadIdx.x] = ls;
  sc[threadIdx.x] = cnt;
  __syncthreads();
  for (int s = 128; s > 0; s >>= 1) {
    if (threadIdx.x < s) {
      sl[threadIdx.x] += sl[threadIdx.x + s];
      sc[threadIdx.x] += sc[threadIdx.x + s];
    }
    __syncthreads();
  }
  if (threadIdx.x == 0) out[0] = (float)(sl[0] / (double)sc[0]);
}

extern "C" void kernel_launch(void* const* d_in, const int* in_sizes, int n_in,
                              void* d_out, int out_size, void* d_ws, size_t ws_size,
                              hipStream_t stream) {
  const float* pred = (const float*)d_in[0];   // (8,19,768,768) f32
  const int*   tgt  = (const int*)d_in[1];     // (8,768,768) i32
  const float* w    = (const float*)d_in[2];   // (19,) f32
  float* out = (float*)d_out;

  char* ws = (char*)d_ws;
  unsigned* ctrl  = (unsigned*)(ws + WS_CTRL);
  unsigned* hist  = (unsigned*)(ws + WS_HIST);
  double*   pLoss = (double*)(ws + WS_PLOSS);
  int*      pCnt  = (int*)(ws + WS_PCNT);
  float*    logp  = (float*)(ws + WS_LOGP);

  hipLaunchKernelGGL(k_init, dim3(32), dim3(256), 0, stream, (unsigned*)ws);

  const int p1_blocks = (int)(NPIX / 16 / 8);  // 16 px/wave, 8 waves/block -> 36864
  hipLaunchKernelGGL(k_pass1, dim3(p1_blocks), dim3(256), 0, stream,
                     pred, tgt, logp, hist, ctrl);
  hipLaunchKernelGGL(k_scan, dim3(1), dim3(1), 0, stream, ctrl, hist, 0);
  for (int lvl = 1; lvl <= 3; ++lvl) {
    hipLaunchKernelGGL(k_rhist, dim3(512), dim3(256), 0, stream,
                       logp, ctrl, hist + 256 * lvl, lvl);
    hipLaunchKernelGGL(k_scan, dim3(1), dim3(1), 0, stream, ctrl, hist + 256 * lvl, lvl);
  }
  hipLaunchKernelGGL(k_final, dim3(FBLOCKS), dim3(256), 0, stream,
                     logp, tgt, w, ctrl, pLoss, pCnt);
  hipLaunchKernelGGL(k_fin, dim3(1), dim3(256), 0, stream, pLoss, pCnt, out);
}